// LSTM_2267742733106
// MI455X (gfx1250) — compile-verified
//
#include <hip/hip_runtime.h>

// ---------------------------------------------------------------------------
// 2-layer LSTM (B=2048, T=1024, D=24, H=8) + Linear(H,24), fp32.
// One wave32 per 16 batch rows; full T-loop fused; V_WMMA_F32_16X16X4_F32
// for the input GEMM (K=24) and the recurrent GEMMs (K=8).
// Biases + activation selection folded into lane-constant FMA coefficients so
// the per-timestep serial chain is: 24 WMMA + 1 exp/rcp per gate value.
// ---------------------------------------------------------------------------

typedef float v2f __attribute__((ext_vector_type(2)));
typedef float v4f __attribute__((ext_vector_type(4)));
typedef float v8f __attribute__((ext_vector_type(8)));

#define B_  2048
#define T_  1024
#define D_  24
#define H_  8
#define O_  24
#define G_  32   // 4*H

#define LOG2E_ 1.4426950408889634f

// generic gated activation: out = om * rcp(1 + exp2(ik*x + ib)) + oa
//  sigmoid(x+b): ik=-log2e, ib=-log2e*b, om=1, oa=0
//  tanh(x+b)   : ik=-2log2e, ib=-2log2e*b, om=2, oa=-1   (tanh = 2*sig(2x)-1)
__device__ __forceinline__ float gate_act(float x, float ik, float ib,
                                          float om, float oa) {
    float e = __builtin_amdgcn_exp2f(__builtin_fmaf(ik, x, ib));
    float r = __builtin_amdgcn_rcpf(1.0f + e);
    return __builtin_fmaf(om, r, oa);
}

__device__ __forceinline__ float tanh_nb(float x) {       // tanh, no bias
    float e = __builtin_amdgcn_exp2f(x * (-2.0f * LOG2E_));
    float r = __builtin_amdgcn_rcpf(1.0f + e);
    return __builtin_fmaf(2.0f, r, -1.0f);
}

__global__ __launch_bounds__(32)
void lstm2_fused(const float* __restrict__ x,
                 const float* __restrict__ w_ih0, const float* __restrict__ w_hh0,
                 const float* __restrict__ b_ih0, const float* __restrict__ b_hh0,
                 const float* __restrict__ w_ih1, const float* __restrict__ w_hh1,
                 const float* __restrict__ b_ih1, const float* __restrict__ b_hh1,
                 const float* __restrict__ w_lin, const float* __restrict__ b_lin,
                 float* __restrict__ out)
{
    __shared__ __align__(16) float gatesLDS[16 * G_];  // [b][g] activated gates
    __shared__ __align__(16) float h0LDS[16 * H_];     // [b][j]
    __shared__ __align__(16) float h1LDS[16 * H_];

    const int lane = threadIdx.x;       // 0..31 (wave32)
    const int n    = lane & 15;         // column (gate) / row (batch) index
    const int half = lane >> 4;         // lane half selects K-pair / M-half
    const int b0   = blockIdx.x * 16;   // batch tile base

    // ---- preload weights into WMMA B-matrix layout -------------------------
    // B (KxN) for 16x16x4 f32: lane<16 holds {K=4kk, 4kk+1}, lane>=16 {4kk+2, 4kk+3},
    // column N = lane%16.  B[k][nn] = W[nt*16+nn][k]   (gates = act @ W^T)
    v2f Bx[6][2];
#pragma unroll
    for (int kk = 0; kk < 6; ++kk)
#pragma unroll
        for (int nt = 0; nt < 2; ++nt) {
            int g = nt * 16 + n;
            int k = kk * 4 + half * 2;
            Bx[kk][nt] = *(const v2f*)(w_ih0 + g * D_ + k);
        }
    v2f Bh0[2][2], Bi1[2][2], Bh1[2][2];
#pragma unroll
    for (int kk = 0; kk < 2; ++kk)
#pragma unroll
        for (int nt = 0; nt < 2; ++nt) {
            int g = nt * 16 + n;
            int k = kk * 4 + half * 2;
            Bh0[kk][nt] = *(const v2f*)(w_hh0 + g * H_ + k);
            Bi1[kk][nt] = *(const v2f*)(w_ih1 + g * H_ + k);
            Bh1[kk][nt] = *(const v2f*)(w_hh1 + g * H_ + k);
        }

    // ---- lane-constant activation coefficients (bias folded in) ------------
    // tile0 columns are gates 0..15 (i,f): always sigmoid.
    // tile1 columns are gates 16..31 (g for n<8: tanh; o for n>=8: sigmoid).
    float bias0t0, bias0t1, bias1t0, bias1t1;
    {
        int g0 = n, g1 = 16 + n;
        bias0t0 = b_ih0[g0] + b_hh0[g0];
        bias0t1 = b_ih0[g1] + b_hh0[g1];
        bias1t0 = b_ih1[g0] + b_hh1[g0];
        bias1t1 = b_ih1[g1] + b_hh1[g1];
    }
    const float ik0 = -LOG2E_;                               // tile0: sigmoid
    const float ib0_l0 = ik0 * bias0t0, ib0_l1 = ik0 * bias1t0;
    const float ik1 = (n < 8) ? (-2.0f * LOG2E_) : -LOG2E_;  // tile1: tanh|sigmoid
    const float ib1_l0 = ik1 * bias0t1, ib1_l1 = ik1 * bias1t1;
    const float om1 = (n < 8) ? 2.0f : 1.0f;
    const float oa1 = (n < 8) ? -1.0f : 0.0f;

    // per-lane x pointer, already offset into A-matrix layout
    const float* xlane = x + (size_t)(b0 + n) * T_ * D_ + half * 2;

    // hidden state in A-matrix layout (two K-chunks each), cell state per lane
    v2f h0a0 = {0.f, 0.f}, h0a1 = {0.f, 0.f};
    v2f h1a0 = {0.f, 0.f}, h1a1 = {0.f, 0.f};
    v4f c0 = {0.f, 0.f, 0.f, 0.f}, c1 = {0.f, 0.f, 0.f, 0.f};

    const int bb = n;              // batch row this lane updates elementwise
    const int j0 = half * 4;       // hidden slice j0..j0+3
    const int brow = half * 8;     // D layout: VGPR v holds M=v (lanes<16) / v+8

    v2f xa[6];
#pragma unroll
    for (int kk = 0; kk < 6; ++kk) xa[kk] = *(const v2f*)(xlane + kk * 4);

    for (int t = 0; t < T_; ++t) {
        // ================= layer 0 gates =================
        v8f acc0 = {0.f, 0.f, 0.f, 0.f, 0.f, 0.f, 0.f, 0.f};
        v8f acc1 = {0.f, 0.f, 0.f, 0.f, 0.f, 0.f, 0.f, 0.f};

#pragma unroll
        for (int kk = 0; kk < 6; ++kk) {
            acc0 = __builtin_amdgcn_wmma_f32_16x16x4_f32(false, xa[kk], false, Bx[kk][0], (short)0, acc0, false, false);
            acc1 = __builtin_amdgcn_wmma_f32_16x16x4_f32(false, xa[kk], false, Bx[kk][1], (short)0, acc1, false, false);
        }
        acc0 = __builtin_amdgcn_wmma_f32_16x16x4_f32(false, h0a0, false, Bh0[0][0], (short)0, acc0, false, false);
        acc1 = __builtin_amdgcn_wmma_f32_16x16x4_f32(false, h0a0, false, Bh0[0][1], (short)0, acc1, false, false);
        acc0 = __builtin_amdgcn_wmma_f32_16x16x4_f32(false, h0a1, false, Bh0[1][0], (short)0, acc0, false, false);
        acc1 = __builtin_amdgcn_wmma_f32_16x16x4_f32(false, h0a1, false, Bh0[1][1], (short)0, acc1, false, false);

        // prefetch next timestep's x while the elementwise phase runs
        {
            int tn = (t + 1 < T_) ? (t + 1) : t;
            const float* xp = xlane + (size_t)tn * D_;
#pragma unroll
            for (int kk = 0; kk < 6; ++kk) xa[kk] = *(const v2f*)(xp + kk * 4);
        }

        // activations (bias folded) in D layout, scatter to LDS [b][g]
#pragma unroll
        for (int v = 0; v < 8; ++v) {
            float s0 = gate_act(acc0[v], ik0, ib0_l0, 1.0f, 0.0f);
            float s1 = gate_act(acc1[v], ik1, ib1_l0, om1, oa1);
            gatesLDS[(brow + v) * G_ + n]      = s0;
            gatesLDS[(brow + v) * G_ + 16 + n] = s1;
        }
        __syncthreads();

        // LSTM cell update, lane owns (bb, j0..j0+3)
        {
            const float* gp = gatesLDS + bb * G_;
            v4f ig = *(const v4f*)(gp + 0  + j0);
            v4f fg = *(const v4f*)(gp + 8  + j0);
            v4f gg = *(const v4f*)(gp + 16 + j0);
            v4f og = *(const v4f*)(gp + 24 + j0);
            v4f hh;
#pragma unroll
            for (int k = 0; k < 4; ++k) {
                c0[k] = __builtin_fmaf(fg[k], c0[k], ig[k] * gg[k]);
                hh[k] = og[k] * tanh_nb(c0[k]);
            }
            *(v4f*)(h0LDS + bb * H_ + j0) = hh;
        }
        __syncthreads();

        // re-read h0 in A-matrix layout
        h0a0 = *(const v2f*)(h0LDS + n * H_ + half * 2);
        h0a1 = *(const v2f*)(h0LDS + n * H_ + 4 + half * 2);

        // ================= layer 1 gates =================
        acc0 = (v8f){0.f, 0.f, 0.f, 0.f, 0.f, 0.f, 0.f, 0.f};
        acc1 = (v8f){0.f, 0.f, 0.f, 0.f, 0.f, 0.f, 0.f, 0.f};

        acc0 = __builtin_amdgcn_wmma_f32_16x16x4_f32(false, h0a0, false, Bi1[0][0], (short)0, acc0, false, false);
        acc1 = __builtin_amdgcn_wmma_f32_16x16x4_f32(false, h0a0, false, Bi1[0][1], (short)0, acc1, false, false);
        acc0 = __builtin_amdgcn_wmma_f32_16x16x4_f32(false, h0a1, false, Bi1[1][0], (short)0, acc0, false, false);
        acc1 = __builtin_amdgcn_wmma_f32_16x16x4_f32(false, h0a1, false, Bi1[1][1], (short)0, acc1, false, false);
        acc0 = __builtin_amdgcn_wmma_f32_16x16x4_f32(false, h1a0, false, Bh1[0][0], (short)0, acc0, false, false);
        acc1 = __builtin_amdgcn_wmma_f32_16x16x4_f32(false, h1a0, false, Bh1[0][1], (short)0, acc1, false, false);
        acc0 = __builtin_amdgcn_wmma_f32_16x16x4_f32(false, h1a1, false, Bh1[1][0], (short)0, acc0, false, false);
        acc1 = __builtin_amdgcn_wmma_f32_16x16x4_f32(false, h1a1, false, Bh1[1][1], (short)0, acc1, false, false);

#pragma unroll
        for (int v = 0; v < 8; ++v) {
            float s0 = gate_act(acc0[v], ik0, ib0_l1, 1.0f, 0.0f);
            float s1 = gate_act(acc1[v], ik1, ib1_l1, om1, oa1);
            gatesLDS[(brow + v) * G_ + n]      = s0;
            gatesLDS[(brow + v) * G_ + 16 + n] = s1;
        }
        __syncthreads();

        {
            const float* gp = gatesLDS + bb * G_;
            v4f ig = *(const v4f*)(gp + 0  + j0);
            v4f fg = *(const v4f*)(gp + 8  + j0);
            v4f gg = *(const v4f*)(gp + 16 + j0);
            v4f og = *(const v4f*)(gp + 24 + j0);
            v4f hh;
#pragma unroll
            for (int k = 0; k < 4; ++k) {
                c1[k] = __builtin_fmaf(fg[k], c1[k], ig[k] * gg[k]);
                hh[k] = og[k] * tanh_nb(c1[k]);
            }
            *(v4f*)(h1LDS + bb * H_ + j0) = hh;
        }
        __syncthreads();

        h1a0 = *(const v2f*)(h1LDS + n * H_ + half * 2);
        h1a1 = *(const v2f*)(h1LDS + n * H_ + 4 + half * 2);
    }

    // ---- final projection: out[b] = h1_T @ w_lin^T + b_lin (tiny, scalar) ----
    {
        const int obase = half * 12;            // lane covers 12 of 24 outputs
        const float* hp = h1LDS + bb * H_;
#pragma unroll
        for (int oo = 0; oo < 12; ++oo) {
            int o = obase + oo;
            float s = b_lin[o];
#pragma unroll
            for (int j = 0; j < H_; ++j) s += hp[j] * w_lin[o * H_ + j];
            out[(size_t)(b0 + bb) * O_ + o] = s;
        }
    }
}

extern "C" void kernel_launch(void* const* d_in, const int* in_sizes, int n_in,
                              void* d_out, int out_size, void* d_ws, size_t ws_size,
                              hipStream_t stream) {
    (void)in_sizes; (void)n_in; (void)d_ws; (void)ws_size; (void)out_size;
    const float* x     = (const float*)d_in[0];
    const float* w_ih0 = (const float*)d_in[1];
    const float* w_hh0 = (const float*)d_in[2];
    const float* b_ih0 = (const float*)d_in[3];
    const float* b_hh0 = (const float*)d_in[4];
    const float* w_ih1 = (const float*)d_in[5];
    const float* w_hh1 = (const float*)d_in[6];
    const float* b_ih1 = (const float*)d_in[7];
    const float* b_hh1 = (const float*)d_in[8];
    const float* w_lin = (const float*)d_in[9];
    const float* b_lin = (const float*)d_in[10];
    float* out = (float*)d_out;

    dim3 grid(B_ / 16);   // 128 independent recurrence chains
    dim3 block(32);       // one wave32 per chain
    hipLaunchKernelGGL(lstm2_fused, grid, block, 0, stream,
                       x, w_ih0, w_hh0, b_ih0, b_hh0,
                       w_ih1, w_hh1, b_ih1, b_hh1, w_lin, b_lin, out);
}